// ModelNew_25056839205238
// MI455X (gfx1250) — compile-verified
//
#include <hip/hip_runtime.h>
#include <stdint.h>

// Reverse (suffix) cumulative sum along rows: out[b, j] = sum_{k>=j} x[b, k]
// B=2048 rows, N=32768 cols, f32.
//
// MI455X plan:
//  * 512 MiB total HBM traffic -> ~23 us floor at 23.3 TB/s. Memory bound.
//  * One row (128 KB) fits in the 320 KB WGP LDS -> one workgroup per row,
//    single pass: async DMA row -> LDS, scan LDS-resident, async DMA -> out.
//  * XOR swizzle on 16B LDS granules kills bank conflicts for both the
//    lane-contiguous copy pattern and the 256B-strided per-thread chunks.
//  * Cross-wave offset scan done with v_wmma_f32_16x16x4_f32 (triangular
//    ones matrix x wave-totals), 4 chained WMMAs.

#define ROW_N   32768
#define BLOCK_T 512
#define PER_T   (ROW_N / BLOCK_T)   // 64 elements per thread
#define QPT     (PER_T / 4)         // 16 16-byte quads per thread
#define ROW_Q   (ROW_N / 4)         // 8192 quads per row
#define ITER1   (ROW_Q / BLOCK_T)   // 16 copy iterations per thread

typedef __attribute__((ext_vector_type(2))) float v2f;
typedef __attribute__((ext_vector_type(4))) float v4f;
typedef __attribute__((ext_vector_type(8))) float v8f;

// XOR swizzle on 16B granule index: column (bits 3:0) ^ row (bits 7:4).
// Copy phase (g = i*512 + t) and chunk phase (g = t*16 + i) both become
// conflict-minimal (2-way over 32 lanes = LDS b128 minimum).
__device__ __forceinline__ int swz(int g) { return g ^ ((g >> 4) & 0xF); }

__global__ void __launch_bounds__(BLOCK_T)
suffix_cumsum_kernel(const float* __restrict__ x, float* __restrict__ out) {
  extern __shared__ __align__(16) float smem[];  // ROW_N floats, swizzled
  __shared__ float s_wtot[16];                   // per-wave totals

  const int t    = threadIdx.x;
  const int lane = t & 31;
  const int wave = t >> 5;
  const int64_t rowbase = (int64_t)blockIdx.x * ROW_N;
  const float* xr  = x   + rowbase;
  float*       orw = out + rowbase;

  // ---- Phase 1: stage whole 128 KB row into LDS (coalesced, async) ----
#if defined(__gfx1250__)
#pragma unroll
  for (int i = 0; i < ITER1; ++i) {
    int g = i * BLOCK_T + t;                       // quad index in row
    // generic LDS pointer low 32 bits == LDS byte offset
    uint32_t l = (uint32_t)(uintptr_t)smem + (uint32_t)(swz(g) * 16);
    const float* gp = xr + g * 4;
    asm volatile("global_load_async_to_lds_b128 %0, %1, off"
                 :: "v"(l), "v"(gp) : "memory");
  }
  asm volatile("s_wait_asynccnt 0x0" ::: "memory");
#else
  for (int i = 0; i < ITER1; ++i) {
    int g = i * BLOCK_T + t;
    v4f q = *(const v4f*)(xr + g * 4);
    *(v4f*)(smem + swz(g) * 4) = q;
  }
#endif
  __syncthreads();

  // ---- Phase 2: per-thread chunk of 64 contiguous elements, local suffix scan ----
  float v[PER_T];
#pragma unroll
  for (int i = 0; i < QPT; ++i) {
    int g = t * QPT + i;
    v4f q = *(const v4f*)(smem + swz(g) * 4);
    v[i * 4 + 0] = q.x; v[i * 4 + 1] = q.y;
    v[i * 4 + 2] = q.z; v[i * 4 + 3] = q.w;
  }
#pragma unroll
  for (int j = PER_T - 2; j >= 0; --j) v[j] += v[j + 1];
  const float ctot = v[0];  // chunk total

  // ---- Phase 3: suffix scan of chunk totals across the wave (32 lanes) ----
  float incl = ctot;
#pragma unroll
  for (int off = 1; off < 32; off <<= 1) {
    float o = __shfl_down(incl, off, 32);
    if (lane + off < 32) incl += o;
  }
  const float exclWave = incl - ctot;          // totals of later lanes in wave
  const float waveTot  = __shfl(incl, 0, 32);  // total of this wave
  if (lane == 0) s_wtot[wave] = waveTot;
  __syncthreads();

  // ---- Phase 3b: exclusive suffix scan of the 16 wave totals via WMMA ----
  // y = H * W,  H[m,k] = (k > m) ? 1 : 0 (strictly upper-tri of the suffix),
  // W[k,n] = S[k] for all n. Split K=16 into four K=4 slabs, chain through C.
  // A-layout (f32 16x4): lane l holds row m=l%16; VGPR p, half h=l/16 -> K=4j+2h+p.
  // B-layout (f32 4x16): lane l holds col n=l%16; VGPR p, half h -> K=4j+2h+p.
  float yw;
#if defined(__gfx1250__) && __has_builtin(__builtin_amdgcn_wmma_f32_16x16x4_f32)
  {
    const int m = lane & 15;
    const int h = lane >> 4;
    v8f acc = {0.f, 0.f, 0.f, 0.f, 0.f, 0.f, 0.f, 0.f};
#pragma unroll
    for (int j = 0; j < 4; ++j) {
      const int k0 = 4 * j + 2 * h;
      const int k1 = k0 + 1;
      v2f a, b;
      a.x = (k0 > m) ? 1.0f : 0.0f;
      a.y = (k1 > m) ? 1.0f : 0.0f;
      b.x = s_wtot[k0];
      b.y = s_wtot[k1];
      acc = __builtin_amdgcn_wmma_f32_16x16x4_f32(
          /*neg_a=*/false, a, /*neg_b=*/false, b,
          /*c_mod=*/(short)0, acc, /*reuse_a=*/false, /*reuse_b=*/false);
    }
    // D-layout: lane l holds y[m'] for m' = j + 8*(l/16), j = VGPR 0..7.
    // All 16 columns are identical, so any lane in the right half has y[wave].
    float mine = acc[wave & 7];
    yw = __shfl(mine, (wave >> 3) << 4, 32);  // lane 0 (waves 0-7) or 16 (8-15)
  }
#else
  {
    float s = 0.f;
    for (int k = wave + 1; k < 16; ++k) s += s_wtot[k];
    yw = s;
  }
#endif

  // ---- Phase 4: apply offsets, write results back to LDS (swizzled) ----
  const float offs = exclWave + yw;
#pragma unroll
  for (int i = 0; i < QPT; ++i) {
    int g = t * QPT + i;
    v4f q;
    q.x = v[i * 4 + 0] + offs; q.y = v[i * 4 + 1] + offs;
    q.z = v[i * 4 + 2] + offs; q.w = v[i * 4 + 3] + offs;
    *(v4f*)(smem + swz(g) * 4) = q;
  }
  __syncthreads();

  // ---- Phase 5: stream row back to HBM (coalesced, async from LDS) ----
#if defined(__gfx1250__)
#pragma unroll
  for (int i = 0; i < ITER1; ++i) {
    int g = i * BLOCK_T + t;
    uint32_t l = (uint32_t)(uintptr_t)smem + (uint32_t)(swz(g) * 16);
    float* gp = orw + g * 4;
    asm volatile("global_store_async_from_lds_b128 %0, %1, off"
                 :: "v"(gp), "v"(l) : "memory");
  }
  asm volatile("s_wait_asynccnt 0x0" ::: "memory");
#else
  for (int i = 0; i < ITER1; ++i) {
    int g = i * BLOCK_T + t;
    *(v4f*)(orw + g * 4) = *(const v4f*)(smem + swz(g) * 4);
  }
#endif
}

extern "C" void kernel_launch(void* const* d_in, const int* in_sizes, int n_in,
                              void* d_out, int out_size, void* d_ws, size_t ws_size,
                              hipStream_t stream) {
  const float* x = (const float*)d_in[0];
  float* out = (float*)d_out;
  (void)d_ws; (void)ws_size; (void)n_in; (void)out_size;

  const int rows = in_sizes[0] / ROW_N;               // 2048
  const size_t smem_bytes = (size_t)ROW_N * sizeof(float);  // 128 KB / WG

  // Allow >64KB dynamic LDS (MI455X WGP has 320 KB).
  hipFuncSetAttribute(reinterpret_cast<const void*>(suffix_cumsum_kernel),
                      hipFuncAttributeMaxDynamicSharedMemorySize,
                      (int)smem_bytes);

  suffix_cumsum_kernel<<<dim3(rows), dim3(BLOCK_T), smem_bytes, stream>>>(x, out);
}